// VCRender_42872363549158
// MI455X (gfx1250) — compile-verified
//
#include <hip/hip_runtime.h>
#include <math.h>

#define HGT 224
#define WID 224
#define NB  2
#define NP  2048
#define NF  256
#define SIGMA_F 0.0003f
#define EPSV 1e-10f

#define GEOM_STRIDE 32
#define COL_STRIDE  12
#define XT_PER_ROW (WID / 16)   // 14

typedef float v2f __attribute__((ext_vector_type(2)));
typedef float v4f __attribute__((ext_vector_type(4)));
typedef float v8f __attribute__((ext_vector_type(8)));

// ---------------------------------------------------------------------------
// Kernel 1: per (batch, face) setup. 512 tasks.
// ---------------------------------------------------------------------------
__global__ void vcr_prep(const float* __restrict__ pts,
                         const int*   __restrict__ faces,
                         const float* __restrict__ rot,
                         const float* __restrict__ pos,
                         const float* __restrict__ proj,
                         const float* __restrict__ cols,
                         float* __restrict__ geom,
                         float* __restrict__ fcol,
                         float* __restrict__ outNormal) {
  int task = blockIdx.x * blockDim.x + threadIdx.x;
  if (task >= NB * NF) return;
  int b = task / NF;
  int f = task - b * NF;

  float R[9], T[3];
#pragma unroll
  for (int k = 0; k < 9; ++k) R[k] = rot[b * 9 + k];
#pragma unroll
  for (int k = 0; k < 3; ++k) T[k] = pos[b * 3 + k];
  float pj0 = proj[0], pj1 = proj[1], pj2 = proj[2];

  float pc[3][3], qx[3], qy[3];
#pragma unroll
  for (int v = 0; v < 3; ++v) {
    int pi = faces[f * 3 + v];
    float w0 = pts[(b * NP + pi) * 3 + 0] - T[0];
    float w1 = pts[(b * NP + pi) * 3 + 1] - T[1];
    float w2 = pts[(b * NP + pi) * 3 + 2] - T[2];
#pragma unroll
    for (int i = 0; i < 3; ++i)
      pc[v][i] = w0 * R[i * 3 + 0] + w1 * R[i * 3 + 1] + w2 * R[i * 3 + 2];
    float zz = pc[v][2] * pj2;
    qx[v] = pc[v][0] * pj0 / zz;
    qy[v] = pc[v][1] * pj1 / zz;
#pragma unroll
    for (int ch = 0; ch < 3; ++ch)
      fcol[(b * NF + f) * COL_STRIDE + v * 3 + ch] = cols[(b * NP + pi) * 3 + ch];
  }

  // normal = cross(p1-p0, p2-p0), normalized with +EPS under sqrt
  float e1x = pc[1][0] - pc[0][0], e1y = pc[1][1] - pc[0][1], e1z = pc[1][2] - pc[0][2];
  float e2x = pc[2][0] - pc[0][0], e2y = pc[2][1] - pc[0][1], e2z = pc[2][2] - pc[0][2];
  float nx = e1y * e2z - e1z * e2y;
  float ny = e1z * e2x - e1x * e2z;
  float nz = e1x * e2y - e1y * e2x;
  float ninv = 1.0f / sqrtf(nx * nx + ny * ny + nz * nz + EPSV);
  outNormal[(b * NF + f) * 3 + 0] = nx * ninv;
  outNormal[(b * NF + f) * 3 + 1] = ny * ninv;
  outNormal[(b * NF + f) * 3 + 2] = nz * ninv;

  float x0 = qx[0], y0 = qy[0], x1 = qx[1], y1 = qy[1], x2 = qx[2], y2 = qy[2];
  float denom = (y1 - y2) * (x0 - x2) + (x2 - x1) * (y0 - y2);
  if (fabsf(denom) < EPSV) denom = EPSV;
  float rd = 1.0f / denom;
  float A0 = (y1 - y2) * rd, B0 = (x2 - x1) * rd;
  float C0 = (-(y1 - y2) * x2 - (x2 - x1) * y2) * rd;
  float A1 = (y2 - y0) * rd, B1 = (x0 - x2) * rd;
  float C1 = (-(y2 - y0) * x2 - (x0 - x2) * y2) * rd;
  float dz0 = pc[0][2] - pc[2][2];
  float dz1 = pc[1][2] - pc[2][2];
  float Az = A0 * dz0 + A1 * dz1;
  float Bz = B0 * dz0 + B1 * dz1;
  float Cz = C0 * dz0 + C1 * dz1 + pc[2][2];

  float* g = geom + (b * NF + f) * GEOM_STRIDE;
  g[0] = A0; g[1] = B0; g[2] = C0;
  g[3] = A1; g[4] = B1; g[5] = C1;
  g[6] = Az; g[7] = Bz; g[8] = Cz;
  g[9] = (nz > 0.0f) ? 1.0f : 0.0f;
  g[10] = x0; g[11] = y0; g[12] = x1; g[13] = y1; g[14] = x2; g[15] = y2;
  float ex, ey;
  ex = x1 - x0; ey = y1 - y0;
  g[16] = ex; g[17] = ey; g[18] = 1.0f / (ex * ex + ey * ey + EPSV);
  ex = x2 - x1; ey = y2 - y1;
  g[19] = ex; g[20] = ey; g[21] = 1.0f / (ex * ex + ey * ey + EPSV);
  ex = x0 - x2; ey = y0 - y2;
  g[22] = ex; g[23] = ey; g[24] = 1.0f / (ex * ex + ey * ey + EPSV);
  g[25] = 0.0f; g[26] = 0.0f; g[27] = 0.0f;
  g[28] = 0.0f; g[29] = 0.0f; g[30] = 0.0f; g[31] = 0.0f;
}

__device__ __forceinline__ float seg_d2(float px, float py, float ax, float ay,
                                        float ex, float ey, float inv) {
  float pax = px - ax, pay = py - ay;
  float t = (pax * ex + pay * ey) * inv;
  t = fminf(fmaxf(t, 0.0f), 1.0f);
  float dx = pax - t * ex, dy = pay - t * ey;
  return dx * dx + dy * dy;
}

// ---------------------------------------------------------------------------
// Kernel 2: rasterize. One wave32 = 16 pixels (one row segment) x 256 faces.
// Per 16-face chunk: 3x V_WMMA_F32_16X16X4_F32 evaluate lam0/lam1/zpix for
// the full 16x16 pixel-face tile in f32. Branchless VALU tail (cndmask) for
// inside/front tests, depth argmax, edge distances, log1p(-prob) accumulation.
// ---------------------------------------------------------------------------
__global__ void __launch_bounds__(256)
vcr_raster(const float* __restrict__ geom,
           const float* __restrict__ fcol,
           float* __restrict__ out) {
  const int lane = threadIdx.x & 31;
  const int wv   = threadIdx.x >> 5;
  const int tile = blockIdx.x * 8 + wv;            // 0 .. NB*HGT*XT_PER_ROW-1
  const int b  = tile / (HGT * XT_PER_ROW);
  const int r2 = tile - b * (HGT * XT_PER_ROW);
  const int y  = r2 / XT_PER_ROW;
  const int xt = r2 - y * XT_PER_ROW;
  const int half = lane >> 4;
  const int n16  = lane & 15;
  const bool hi  = (half != 0);

  const float step   = 2.0f / (float)WID;
  const float pxBase = (2.0f * (float)(xt * 16) + 1.0f) / (float)WID - 1.0f;
  const float py     = 1.0f - (2.0f * (float)y + 1.0f) / (float)HGT;

  // A matrix (16 pixels x K=4 = [PX, PY, 1, 0]), ISA 32-bit 16x4 layout:
  // lanes 0-15: VGPR0=K0, VGPR1=K1 ; lanes 16-31: VGPR0=K2, VGPR1=K3
  v2f a;
  a.x = hi ? 1.0f : (pxBase + (float)n16 * step);
  a.y = hi ? 0.0f : py;

  float bS[8], bL0[8], bL1[8], lacc[8];
  int   bF[8], vAny[8];
#pragma unroll
  for (int r = 0; r < 8; ++r) {
    bS[r] = -1e10f; bL0[r] = 0.0f; bL1[r] = 0.0f;
    lacc[r] = 0.0f; bF[r] = 0; vAny[r] = 0;
  }

  const v8f cz = {};
  const float* gbase = geom + (b * NF) * GEOM_STRIDE;
  for (int c = 0; c < NF / 16; ++c) {
    const int fi = c * 16 + n16;                    // this lane's face
    const float* g = gbase + fi * GEOM_STRIDE;

    // Unconditional vector loads of the 9 coefficients, then register selects
    // (v_cndmask) to build the B matrices: VGPR r holds K=r (lanes 0-15) and
    // K=r+2 (lanes 16-31); the K=3 row is zero.
    const v4f ga = *(const v4f*)(g);        // A0 B0 C0 A1
    const v4f gb = *(const v4f*)(g + 4);    // B1 C1 Az Bz
    const float gcz = g[8];                 // Cz
    v2f bl0, bl1, bz;
    bl0.x = hi ? ga.z : ga.x;   bl0.y = hi ? 0.0f : ga.y;
    bl1.x = hi ? gb.y : ga.w;   bl1.y = hi ? 0.0f : gb.x;
    bz.x  = hi ? gcz  : gb.z;   bz.y  = hi ? 0.0f : gb.w;

    v8f L0 = __builtin_amdgcn_wmma_f32_16x16x4_f32(false, a, false, bl0, (short)0, cz, false, false);
    v8f L1 = __builtin_amdgcn_wmma_f32_16x16x4_f32(false, a, false, bl1, (short)0, cz, false, false);
    v8f Z  = __builtin_amdgcn_wmma_f32_16x16x4_f32(false, a, false, bz,  (short)0, cz, false, false);

    const float front = g[9];
    const float x0 = g[10], y0g = g[11], x1 = g[12], y1g = g[13], x2 = g[14], y2g = g[15];
    const float e0x = g[16], e0y = g[17], i0 = g[18];
    const float e1x = g[19], e1y = g[20], i1 = g[21];
    const float e2x = g[22], e2y = g[23], i2 = g[24];

    // Prefetch next chunk's record (wrapped so the address is always valid,
    // avoiding a per-iteration compare/branch). loc=3 -> near-cache prefetch.
    __builtin_prefetch(gbase + ((fi + 16) & (NF - 1)) * GEOM_STRIDE, 0, 3);

    const bool isfront = front > 0.0f;
#pragma unroll
    for (int r = 0; r < 8; ++r) {
      // C/D layout: VGPR r holds pixel M = r + 8*half for this lane
      const float px = pxBase + (float)(r + 8 * half) * step;
      const float l0 = L0[r], l1 = L1[r];
      const float l2 = 1.0f - l0 - l1;
      const bool inside = (l0 >= 0.0f) & (l1 >= 0.0f) & (l2 >= 0.0f);
      const bool valid = inside & isfront;
      const float s = valid ? Z[r] : -1e10f;
      const bool upd = s > bS[r];
      bS[r]  = upd ? s  : bS[r];
      bF[r]  = upd ? fi : bF[r];
      bL0[r] = upd ? l0 : bL0[r];
      bL1[r] = upd ? l1 : bL1[r];
      vAny[r] |= (int)valid;

      float d2 = seg_d2(px, py, x0, y0g, e0x, e0y, i0);
      d2 = fminf(d2, seg_d2(px, py, x1, y1g, e1x, e1y, i1));
      d2 = fminf(d2, seg_d2(px, py, x2, y2g, e2x, e2y, i2));
      d2 = inside ? 0.0f : d2;
      const float p = isfront ? __expf(-d2 * (1.0f / SIGMA_F)) * (1.0f - 1e-7f) : 0.0f;
      lacc[r] += log1pf(-p);
    }
  }

  // Reduce over the 16-lane face group (faces live across lanes mod 16).
#pragma unroll
  for (int m = 1; m < 16; m <<= 1) {
#pragma unroll
    for (int r = 0; r < 8; ++r) {
      const float oS = __shfl_xor(bS[r], m, 32);
      const int   oF = __shfl_xor(bF[r], m, 32);
      const float o0 = __shfl_xor(bL0[r], m, 32);
      const float o1 = __shfl_xor(bL1[r], m, 32);
      const bool take = (oS > bS[r]) || ((oS == bS[r]) && (oF < bF[r]));
      bS[r]  = take ? oS : bS[r];
      bF[r]  = take ? oF : bF[r];
      bL0[r] = take ? o0 : bL0[r];
      bL1[r] = take ? o1 : bL1[r];
      vAny[r] |= __shfl_xor(vAny[r], m, 32);
      lacc[r] += __shfl_xor(lacc[r], m, 32);
    }
  }

  // Writers: lane (n16 == r) of each half writes pixel r + 8*half.
  const int imBase = (b * HGT + y) * WID + xt * 16;
#pragma unroll
  for (int r = 0; r < 8; ++r) {
    if (n16 == r) {
      const int pix = r + 8 * half;
      const int po = imBase + pix;
      const float* cc = fcol + (b * NF + bF[r]) * COL_STRIDE;
      const float l0 = bL0[r], l1 = bL1[r], l2 = 1.0f - l0 - l1;
      const float va = vAny[r] ? 1.0f : 0.0f;
      out[po * 3 + 0] = (l0 * cc[0] + l1 * cc[3] + l2 * cc[6]) * va;
      out[po * 3 + 1] = (l0 * cc[1] + l1 * cc[4] + l2 * cc[7]) * va;
      out[po * 3 + 2] = (l0 * cc[2] + l1 * cc[5] + l2 * cc[8]) * va;
      out[NB * HGT * WID * 3 + po] = 1.0f - __expf(lacc[r]);
    }
  }
}

extern "C" void kernel_launch(void* const* d_in, const int* in_sizes, int n_in,
                              void* d_out, int out_size, void* d_ws, size_t ws_size,
                              hipStream_t stream) {
  (void)in_sizes; (void)n_in; (void)out_size; (void)ws_size;
  const float* pts   = (const float*)d_in[0];  // points_bxpx3 (2,2048,3)
  const int*   faces = (const int*)  d_in[1];  // faces_fx3    (256,3)
  const float* rot   = (const float*)d_in[2];  // camera_rot   (2,3,3)
  const float* pos   = (const float*)d_in[3];  // camera_pos   (2,3)
  const float* proj  = (const float*)d_in[4];  // camera_proj  (3,1)
  const float* cols  = (const float*)d_in[5];  // colors_bxpx3 (2,2048,3)

  float* geom = (float*)d_ws;                         // NB*NF*32 floats
  float* fcol = geom + NB * NF * GEOM_STRIDE;         // NB*NF*12 floats
  float* outf = (float*)d_out;
  float* outNormal = outf + NB * HGT * WID * 3 + NB * HGT * WID;  // offset 401408

  vcr_prep<<<2, 256, 0, stream>>>(pts, faces, rot, pos, proj, cols,
                                  geom, fcol, outNormal);

  const int tiles = NB * HGT * XT_PER_ROW;            // 6272 waves
  vcr_raster<<<tiles / 8, 256, 0, stream>>>(geom, fcol, outf);
}